// IIRDecompositionLayer_4827543241304
// MI455X (gfx1250) — compile-verified
//
#include <hip/hip_runtime.h>

// IIR filtfilt decomposition for MI455X (gfx1250, wave32).
// Forward pass -> w (workspace), reverse pass -> seasonality + trend.
// 16-step time tiles solved as Toeplitz matmul via V_WMMA_F32_16X16X4_F32.
// Staging via GLOBAL_LOAD_ASYNC_TO_LDS_B128 with ping-pong double buffering.
// Wave-uniform task state forced into SGPRs via readfirstlane.

#define BB     16
#define LL     4096
#define DD     512
#define LC     1024                 // chunk length along L
#define HALO   384                  // warmup; pole^384 ~ 4e-16
#define NG     (DD / 16)            // 32 channel groups
#define CHUNKS (LL / LC)            // 4
#define WPB    8                    // waves per 256-thread block
#define TASKS  (BB * CHUNKS * NG)   // 2048 wave-tasks

typedef float v2f __attribute__((ext_vector_type(2)));
typedef float v8f __attribute__((ext_vector_type(8)));

// h[0..16]: impulse response of y[j] = bx[j] - a1 y[j-1] - a2 y[j-2].
// y[j] = sum_k h[j-k] bx[k] + h[j+1]*s1 + (-a2*h[j])*s2, s1=y[-1], s2=y[-2].
__device__ __forceinline__ void build_table(float a1, float a2, float* ldsH, int tid) {
  float hm2 = 1.0f;        // h[0]
  float hm1 = -a1;         // h[1]
  float mine = (tid == 0) ? hm2 : ((tid == 1) ? hm1 : 0.0f);
  for (int j = 2; j <= 16; ++j) {
    float c = -a1 * hm1 - a2 * hm2;
    if (tid == j) mine = c;
    hm2 = hm1; hm1 = c;
  }
  if (tid < 17) ldsH[tid] = mine;
}

// Async-copy 18 staged rows (8 rows per b128 issue; 3 issues) into an LDS buffer.
// Issue k=2 clamps the staged row to 17: lanes qlane>=2 redundantly rewrite row 17
// with identical data (benign), avoiding any exec-mask manipulation.
__device__ __forceinline__ void issue_tile(const float* src, int rowBase,
                                           float* dstBuf, int qlane, int c4) {
  unsigned db = (unsigned)(unsigned long long)dstBuf;
#pragma unroll
  for (int k = 0; k < 3; ++k) {
    int rr = 8 * k + qlane;
    if (k == 2) rr = rr > 17 ? 17 : rr;     // duplicate row 17 for tail lanes
    int row = rowBase + rr;
    row = row < 0 ? 0 : (row > (LL - 1) ? (LL - 1) : row);
    const float* ga = src + (unsigned long long)row * DD + c4;
    unsigned dst = db + (unsigned)((rr * 16 + c4) * 4);
    asm volatile("global_load_async_to_lds_b128 %0, %1, off"
                 :: "v"(dst), "v"(ga) : "memory");
  }
}

// ---------------- forward pass: x -> w ----------------
__global__ void __launch_bounds__(256) iir_fwd(const float* __restrict__ x,
                                               const float* __restrict__ bc,
                                               const float* __restrict__ am,
                                               float* __restrict__ w) {
  __shared__ float ldsH[32];
  __shared__ float stageA[WPB][18 * 16];
  __shared__ float stageB[WPB][18 * 16];

  const int tid = threadIdx.x;
  const float b0 = bc[0], b1 = bc[1], b2 = bc[2];
  const float a1 = -am[0], a2 = -am[1];   // A = [[-a1,-a2],[1,0]]
  build_table(a1, a2, ldsH, tid);
  __syncthreads();

  const int lane  = tid & 31;
  const int hi    = (lane >> 4) & 1;      // half-wave K/M offset selector
  const int n     = lane & 15;            // channel / matrix row
  const int qlane = lane >> 2;            // async-copy row-in-group
  const int c4    = (lane & 3) * 4;       // async-copy channel quad
  const int wv    = tid >> 5;
  float* st0 = stageA[wv];
  float* st1 = stageB[wv];

  // Loop-invariant A fragments: T[m,k] = (m>=k) ? h[m-k] : 0
  v2f afr[4];
#pragma unroll
  for (int f = 0; f < 4; ++f) {
    int k0 = 4 * f + 2 * hi;
    afr[f].x = (n >= k0)     ? ldsH[n - k0]     : 0.0f;
    afr[f].y = (n >= k0 + 1) ? ldsH[n - k0 - 1] : 0.0f;
  }
  float ph[8], qh[8];
#pragma unroll
  for (int r = 0; r < 8; ++r) {
    ph[r] = ldsH[r + 8 * hi + 1];
    qh[r] = -a2 * ldsH[r + 8 * hi];
  }

  // task is wave-uniform by construction; force it scalar so chunk bounds,
  // base offsets and loop control live in SGPRs (s_cmp/s_cbranch, saddr form).
  const int task = __builtin_amdgcn_readfirstlane(blockIdx.x * WPB + wv);
  if (task >= TASKS) return;
  const int bI    = task / (CHUNKS * NG);
  const int rem   = task - bI * (CHUNKS * NG);
  const int chunk = rem / NG;
  const int g     = rem - chunk * NG;
  const size_t baseOff = ((size_t)bI * LL) * DD + (size_t)g * 16;
  const float* xb = x + baseOff;
  float*       wb = w + baseOff;

  const int cs = chunk * LC, ce = cs + LC;
  int t0; float s1, s2;
  if (chunk == 0) {                        // exact init at t=0: y[-1]=y[-2]=bx[0]
    t0 = 0;
    float bs = (b0 + b1 + b2) * xb[n];
    s1 = bs; s2 = bs;
  } else {                                 // zero-state warmup, error ~4e-16
    t0 = cs - HALO;
    s1 = 0.0f; s2 = 0.0f;
  }

  int p = 0;
  issue_tile(xb, t0 - 2, st0, qlane, c4);  // prologue: tile 0 in flight

  auto tileF = [&](int t, bool doStore) {
    float* cur = p ? st1 : st0;
    float* nxt = p ? st0 : st1;
    issue_tile(xb, t + 16 - 2, nxt, qlane, c4);       // next tile (clamped-safe)
    asm volatile("s_wait_asynccnt 0x3" ::: "memory"); // current tile's 3 done
    __builtin_amdgcn_wave_barrier();

    v8f acc;
#pragma unroll
    for (int r = 0; r < 8; ++r) acc[r] = ph[r] * s1 + qh[r] * s2;

#pragma unroll
    for (int f = 0; f < 4; ++f) {
      int i0 = 4 * f + 2 * hi;
      float S0 = cur[(i0    ) * 16 + n];
      float S1 = cur[(i0 + 1) * 16 + n];
      float S2 = cur[(i0 + 2) * 16 + n];
      float S3 = cur[(i0 + 3) * 16 + n];
      v2f bfr;
      bfr.x = b0 * S2 + b1 * S1 + b2 * S0;   // bx[t+i0]
      bfr.y = b0 * S3 + b1 * S2 + b2 * S1;   // bx[t+i0+1]
      acc = __builtin_amdgcn_wmma_f32_16x16x4_f32(false, afr[f], false, bfr,
                                                  (short)0, acc, false, false);
    }
    s1 = __shfl(acc[7], 16 + n, 32);         // y[t+15]
    s2 = __shfl(acc[6], 16 + n, 32);         // y[t+14]

    if (doStore) {
#pragma unroll
      for (int r = 0; r < 8; ++r) {
        int tr = t + r + 8 * hi;
        wb[(size_t)tr * DD + n] = acc[r];
      }
    }
    __builtin_amdgcn_wave_barrier();
    p ^= 1;
  };

  int t = t0;
  for (; t < cs; t += 16) tileF(t, false);   // warmup: no stores, no branches
  for (; t < ce; t += 16) tileF(t, true);    // main: unconditional stores
}

// ---------------- reverse pass: w -> seasonality, trend = x - y ----------------
__global__ void __launch_bounds__(256) iir_rev(const float* __restrict__ x,
                                               const float* __restrict__ w,
                                               const float* __restrict__ bc,
                                               const float* __restrict__ am,
                                               float* __restrict__ season,
                                               float* __restrict__ trend) {
  __shared__ float ldsH[32];
  __shared__ float stageA[WPB][18 * 16];
  __shared__ float stageB[WPB][18 * 16];

  const int tid = threadIdx.x;
  const float b0 = bc[0], b1 = bc[1], b2 = bc[2];
  const float a1 = -am[0], a2 = -am[1];
  build_table(a1, a2, ldsH, tid);
  __syncthreads();

  const int lane  = tid & 31;
  const int hi    = (lane >> 4) & 1;
  const int n     = lane & 15;
  const int qlane = lane >> 2;
  const int c4    = (lane & 3) * 4;
  const int wv    = tid >> 5;
  float* st0 = stageA[wv];
  float* st1 = stageB[wv];

  v2f afr[4];
#pragma unroll
  for (int f = 0; f < 4; ++f) {
    int k0 = 4 * f + 2 * hi;
    afr[f].x = (n >= k0)     ? ldsH[n - k0]     : 0.0f;
    afr[f].y = (n >= k0 + 1) ? ldsH[n - k0 - 1] : 0.0f;
  }
  float ph[8], qh[8];
#pragma unroll
  for (int r = 0; r < 8; ++r) {
    ph[r] = ldsH[r + 8 * hi + 1];
    qh[r] = -a2 * ldsH[r + 8 * hi];
  }

  const int task = __builtin_amdgcn_readfirstlane(blockIdx.x * WPB + wv);
  if (task >= TASKS) return;
  const int bI    = task / (CHUNKS * NG);
  const int rem   = task - bI * (CHUNKS * NG);
  const int chunk = rem / NG;
  const int g     = rem - chunk * NG;
  const size_t baseOff = ((size_t)bI * LL) * DD + (size_t)g * 16;
  const float* xb = x + baseOff;
  const float* wb = w + baseOff;
  float*       sb = season + baseOff;
  float*       tb = trend + baseOff;

  const int cs = chunk * LC, ce = cs + LC;
  int thi; float s1, s2;
  if (ce == LL) {                          // exact init at t=L-1
    thi = LL;
    float bs = (b0 + b1 + b2) * wb[(size_t)(LL - 1) * DD + n];
    s1 = bs; s2 = bs;
  } else {
    thi = ce + HALO;
    s1 = 0.0f; s2 = 0.0f;
  }

  int p = 0;
  issue_tile(wb, thi - 16, st0, qlane, c4);   // rows base..base+17, base=thi-16

  auto tileR = [&](int base, bool doStore) {
    float* cur = p ? st1 : st0;
    float* nxt = p ? st0 : st1;
    issue_tile(wb, base - 16, nxt, qlane, c4);        // next (lower) tile
    asm volatile("s_wait_asynccnt 0x3" ::: "memory");
    __builtin_amdgcn_wave_barrier();

    v8f acc;
#pragma unroll
    for (int r = 0; r < 8; ++r) acc[r] = ph[r] * s1 + qh[r] * s2;

#pragma unroll
    for (int f = 0; f < 4; ++f) {
      int i0 = 4 * f + 2 * hi;
      float Sa = cur[(14 - i0) * 16 + n];
      float Sb = cur[(15 - i0) * 16 + n];
      float Sc = cur[(16 - i0) * 16 + n];
      float Sd = cur[(17 - i0) * 16 + n];
      v2f bfr;
      bfr.x = b0 * Sb + b1 * Sc + b2 * Sd;   // bxr at j=i0   (t=base+15-i0)
      bfr.y = b0 * Sa + b1 * Sb + b2 * Sc;   // bxr at j=i0+1
      acc = __builtin_amdgcn_wmma_f32_16x16x4_f32(false, afr[f], false, bfr,
                                                  (short)0, acc, false, false);
    }
    s1 = __shfl(acc[7], 16 + n, 32);         // y[base]
    s2 = __shfl(acc[6], 16 + n, 32);         // y[base+1]

    if (doStore) {
#pragma unroll
      for (int r = 0; r < 8; ++r) {
        int m  = r + 8 * hi;
        int tr = base + 15 - m;
        size_t off = (size_t)tr * DD + n;
        float yv = acc[r];
        sb[off] = yv;
        tb[off] = xb[off] - yv;
      }
    }
    __builtin_amdgcn_wave_barrier();
    p ^= 1;
  };

  int base = thi - 16;
  for (; base >= ce; base -= 16) tileR(base, false);  // warmup
  for (; base >= cs; base -= 16) tileR(base, true);   // main
}

extern "C" void kernel_launch(void* const* d_in, const int* in_sizes, int n_in,
                              void* d_out, int out_size, void* d_ws, size_t ws_size,
                              hipStream_t stream) {
  const float* x  = (const float*)d_in[0];
  const float* bc = (const float*)d_in[1];   // [1,3]
  const float* am = (const float*)d_in[2];   // [1,2,2] = [[-a1,-a2],[1,0]]
  float* season = (float*)d_out;
  float* trendp = season + (size_t)BB * LL * DD;
  float* w = (float*)d_ws;                   // B*L*D fp32 intermediate

  dim3 blk(256);
  dim3 grd(TASKS / WPB);                     // 256 blocks, 8 wave-tasks each
  iir_fwd<<<grd, blk, 0, stream>>>(x, bc, am, w);
  iir_rev<<<grd, blk, 0, stream>>>(x, w, bc, am, season, trendp);
}